// DynamicOrdering_44693429682798
// MI455X (gfx1250) — compile-verified
//
#include <hip/hip_runtime.h>
#include <stdint.h>

typedef unsigned int u32;

static constexpr int B_    = 32;
static constexpr int N_    = 262144;
static constexpr int TILE  = 1024;          // elements per scatter block (1 per thread, 32 waves)
static constexpr int NBLK  = N_ / TILE;     // 256 blocks per batch
static constexpr int RADIX = 256;

// ---- ordered-uint encoding for float atomic min/max (handles negatives) ----
__device__ __forceinline__ u32 encf(float f) {
  u32 u = __float_as_uint(f);
  return (u & 0x80000000u) ? ~u : (u | 0x80000000u);
}
__device__ __forceinline__ float decf(u32 u) {
  u = (u & 0x80000000u) ? (u & 0x7fffffffu) : ~u;
  return __uint_as_float(u);
}

// ---- wave32 peer matching: mask of lanes in this wave with the same digit ----
__device__ __forceinline__ u32 match_digit(u32 d) {
  u32 peers = 0xffffffffu;
#pragma unroll
  for (int b = 0; b < 8; ++b) {
    bool bit = (d >> b) & 1u;
    u32 m = (u32)__ballot(bit);      // wave32: low 32 bits valid
    peers &= bit ? m : ~m;
  }
  return peers;
}

// ============================ reduction stage ============================

__global__ void init_mm_kernel(u32* __restrict__ mm) {
  int i = threadIdx.x;
  if (i < B_ * 6) mm[i] = ((i % 6) < 3) ? 0xFFFFFFFFu : 0u;  // min slots / max slots
}

__global__ __launch_bounds__(256) void reduce_mm_kernel(const float* __restrict__ x,
                                                        u32* __restrict__ mm) {
  const int b = blockIdx.y;
  const int tid = threadIdx.x;
  const size_t base = (size_t)b * N_;
  float mn0 = 3.4e38f, mn1 = 3.4e38f, mn2 = 3.4e38f;
  float mx0 = -3.4e38f, mx1 = -3.4e38f, mx2 = -3.4e38f;
  const int chunk = blockIdx.x * 4096;
#pragma unroll 4
  for (int it = 0; it < 16; ++it) {
    int i = chunk + it * 256 + tid;
    const float* p = x + (base + (size_t)i) * 3;
    float a0 = p[0], a1 = p[1], a2 = p[2];
    mn0 = fminf(mn0, a0); mx0 = fmaxf(mx0, a0);
    mn1 = fminf(mn1, a1); mx1 = fmaxf(mx1, a1);
    mn2 = fminf(mn2, a2); mx2 = fmaxf(mx2, a2);
  }
#pragma unroll
  for (int off = 16; off > 0; off >>= 1) {
    mn0 = fminf(mn0, __shfl_xor(mn0, off, 32));
    mn1 = fminf(mn1, __shfl_xor(mn1, off, 32));
    mn2 = fminf(mn2, __shfl_xor(mn2, off, 32));
    mx0 = fmaxf(mx0, __shfl_xor(mx0, off, 32));
    mx1 = fmaxf(mx1, __shfl_xor(mx1, off, 32));
    mx2 = fmaxf(mx2, __shfl_xor(mx2, off, 32));
  }
  if ((tid & 31) == 0) {
    atomicMin(&mm[b * 6 + 0], encf(mn0));
    atomicMin(&mm[b * 6 + 1], encf(mn1));
    atomicMin(&mm[b * 6 + 2], encf(mn2));
    atomicMax(&mm[b * 6 + 3], encf(mx0));
    atomicMax(&mm[b * 6 + 4], encf(mx1));
    atomicMax(&mm[b * 6 + 5], encf(mx2));
  }
}

__global__ void params_kernel(const u32* __restrict__ mm, float* __restrict__ prm) {
  int b = threadIdx.x;
  if (b >= B_) return;
  float m0 = decf(mm[b * 6 + 0]);
  float m1 = decf(mm[b * 6 + 1]);
  float m2 = decf(mm[b * 6 + 2]);
  float M = fmaxf(fmaxf(decf(mm[b * 6 + 3]) - m0, decf(mm[b * 6 + 4]) - m1),
                  decf(mm[b * 6 + 5]) - m2);
  prm[b * 4 + 0] = m0;
  prm[b * 4 + 1] = m1;
  prm[b * 4 + 2] = m2;
  prm[b * 4 + 3] = fmaxf(M, 1e-8f);  // denominator, division done per element like the reference
}

// ============================ histogram stage ============================

__global__ __launch_bounds__(256) void hist_first_kernel(const float* __restrict__ x,
                                                         const float* __restrict__ prm,
                                                         u32* __restrict__ counts) {
  __shared__ u32 h[RADIX];
  const int tid = threadIdx.x, blk = blockIdx.x, b = blockIdx.y;
  h[tid] = 0;
  __syncthreads();
  const float m2 = prm[b * 4 + 2], den = prm[b * 4 + 3];
  const size_t base = (size_t)b * N_;
#pragma unroll
  for (int it = 0; it < TILE / 256; ++it) {
    int i = blk * TILE + it * 256 + tid;
    float v = x[(base + (size_t)i) * 3 + 2];
    u32 k = __float_as_uint((v - m2) / den);   // pass 0 digit = byte0 of z-key
    atomicAdd(&h[k & 0xFFu], 1u);
  }
  __syncthreads();
  counts[((size_t)b * RADIX + tid) * NBLK + blk] = h[tid];
}

__global__ __launch_bounds__(256) void hist_mid_kernel(const u32* __restrict__ srcW, int shift,
                                                       u32* __restrict__ counts) {
  __shared__ u32 h[RADIX];
  const int tid = threadIdx.x, blk = blockIdx.x, b = blockIdx.y;
  h[tid] = 0;
  __syncthreads();
  // one b128 load per thread covers the 1024-element tile
  const uint4* src4 = (const uint4*)(srcW + (size_t)b * N_);
  uint4 v = src4[(size_t)blk * (TILE / 4) + tid];
  atomicAdd(&h[(v.x >> shift) & 0xFFu], 1u);
  atomicAdd(&h[(v.y >> shift) & 0xFFu], 1u);
  atomicAdd(&h[(v.z >> shift) & 0xFFu], 1u);
  atomicAdd(&h[(v.w >> shift) & 0xFFu], 1u);
  __syncthreads();
  counts[((size_t)b * RADIX + tid) * NBLK + blk] = h[tid];
}

// per-batch: digit totals -> exclusive digit base -> in-place per-(digit,block) offsets
__global__ __launch_bounds__(256) void scan_kernel(u32* __restrict__ counts) {
  const int b = blockIdx.x, d = threadIdx.x;
  u32* c = counts + ((size_t)b * RADIX + d) * NBLK;
  u32 sum = 0;
  for (int k = 0; k < NBLK; ++k) sum += c[k];
  __shared__ u32 sc[RADIX];
  sc[d] = sum;
  __syncthreads();
  u32 v = sum;
  for (int off = 1; off < RADIX; off <<= 1) {
    u32 t = (d >= off) ? sc[d - off] : 0u;
    __syncthreads();
    v += t;
    sc[d] = v;
    __syncthreads();
  }
  u32 run = v - sum;  // exclusive base of digit d within batch
  for (int k = 0; k < NBLK; ++k) {
    u32 cc = c[k];
    c[k] = run;
    run += cc;
  }
}

// ============================ scatter stage ============================

// shared ranking logic: returns global position for this element
__device__ __forceinline__ u32 stable_rank(u32 d, u32* wh, const u32* __restrict__ counts,
                                           int b, int blk, int tid) {
  const int lane = tid & 31, wv = tid >> 5;
  u32 peers = match_digit(d);
  u32 rank = __popc(peers & ((1u << lane) - 1u));
  __syncthreads();  // wh zero-init complete
  if (lane == __builtin_ctz(peers)) wh[wv * RADIX + d] = __popc(peers);
  __syncthreads();
  if (tid < RADIX) {  // exclusive scan over 32 waves per digit
    u32 run = 0;
    for (int w = 0; w < 32; ++w) {
      u32 c = wh[w * RADIX + tid];
      wh[w * RADIX + tid] = run;
      run += c;
    }
  }
  __syncthreads();
  return counts[((size_t)b * RADIX + d) * NBLK + blk] + wh[wv * RADIX + d] + rank;
}

// stage the tile's three key words into LDS via CDNA5 async DMA (B128, lanes 0..255)
__device__ __forceinline__ void stage_tile_async(const u32* p0, const u32* p1, const u32* p2,
                                                 int blk, int tid) {
  if (tid < TILE / 4) {
    u32 goff = (u32)(blk * TILE * 4 + tid * 16);  // byte offset into each SoA array
    u32 l0 = (u32)(tid * 16);
    u32 l1 = 4096u + (u32)(tid * 16);
    u32 l2 = 8192u + (u32)(tid * 16);
    asm volatile("global_load_async_to_lds_b128 %0, %1, %2" ::"v"(l0), "v"(goff), "s"(p0)
                 : "memory");
    asm volatile("global_load_async_to_lds_b128 %0, %1, %2" ::"v"(l1), "v"(goff), "s"(p1)
                 : "memory");
    asm volatile("global_load_async_to_lds_b128 %0, %1, %2" ::"v"(l2), "v"(goff), "s"(p2)
                 : "memory");
  }
}

// pass 0: build keys from x, scatter by byte0 of z-key
__global__ __launch_bounds__(1024) void scatter_first_kernel(
    const float* __restrict__ x, const float* __restrict__ prm, const u32* __restrict__ counts,
    u32* __restrict__ d0, u32* __restrict__ d1, u32* __restrict__ d2) {
  extern __shared__ u32 smem[];
  u32* wh = smem;  // [32][256]
  const int tid = threadIdx.x, blk = blockIdx.x, b = blockIdx.y;
  for (int j = tid; j < 32 * RADIX; j += 1024) wh[j] = 0;
  const size_t base = (size_t)b * N_;
  const int i = blk * TILE + tid;
  const float* p = x + (base + (size_t)i) * 3;
  const float m0 = prm[b * 4 + 0], m1 = prm[b * 4 + 1], m2 = prm[b * 4 + 2], den = prm[b * 4 + 3];
  u32 k0 = __float_as_uint((p[0] - m0) / den);
  u32 k1 = __float_as_uint((p[1] - m1) / den);
  u32 k2 = __float_as_uint((p[2] - m2) / den);
  u32 pos = stable_rank(k2 & 0xFFu, wh, counts, b, blk, tid);
  d0[base + pos] = k0;
  d1[base + pos] = k1;
  d2[base + pos] = k2;
}

// passes 1..10: async-DMA tile into LDS (ASYNCcnt path), rank, scatter SoA
__global__ __launch_bounds__(1024) void scatter_mid_kernel(
    const u32* __restrict__ s0, const u32* __restrict__ s1, const u32* __restrict__ s2,
    u32* __restrict__ d0, u32* __restrict__ d1, u32* __restrict__ d2,
    int wsel, int shift, const u32* __restrict__ counts) {
  extern __shared__ u32 smem[];
  u32* st = smem;             // staging: 3 * 1024 u32, LDS byte offsets 0 / 4096 / 8192
  u32* wh = smem + 3 * TILE;  // [32][256]
  const int tid = threadIdx.x, blk = blockIdx.x, b = blockIdx.y;
  const size_t base = (size_t)b * N_;
  stage_tile_async(s0 + base, s1 + base, s2 + base, blk, tid);
  for (int j = tid; j < 32 * RADIX; j += 1024) wh[j] = 0;  // overlap with DMA
  asm volatile("s_wait_asynccnt 0x0" ::: "memory");
  __syncthreads();
  u32 k0 = st[tid], k1 = st[TILE + tid], k2 = st[2 * TILE + tid];
  u32 key = (wsel == 0) ? k0 : ((wsel == 1) ? k1 : k2);
  u32 pos = stable_rank((key >> shift) & 0xFFu, wh, counts, b, blk, tid);
  d0[base + pos] = k0;
  d1[base + pos] = k1;
  d2[base + pos] = k2;
}

// pass 11: digit = byte3 of x-key; write interleaved (B,N,3) floats directly
__global__ __launch_bounds__(1024) void scatter_last_kernel(
    const u32* __restrict__ s0, const u32* __restrict__ s1, const u32* __restrict__ s2,
    float* __restrict__ out, const u32* __restrict__ counts) {
  extern __shared__ u32 smem[];
  u32* st = smem;
  u32* wh = smem + 3 * TILE;
  const int tid = threadIdx.x, blk = blockIdx.x, b = blockIdx.y;
  const size_t base = (size_t)b * N_;
  stage_tile_async(s0 + base, s1 + base, s2 + base, blk, tid);
  for (int j = tid; j < 32 * RADIX; j += 1024) wh[j] = 0;
  asm volatile("s_wait_asynccnt 0x0" ::: "memory");
  __syncthreads();
  u32 k0 = st[tid], k1 = st[TILE + tid], k2 = st[2 * TILE + tid];
  u32 pos = stable_rank((k0 >> 24) & 0xFFu, wh, counts, b, blk, tid);
  float* o = out + (base + pos) * 3;
  o[0] = __uint_as_float(k0);   // sorted keys ARE the normalized coordinates
  o[1] = __uint_as_float(k1);
  o[2] = __uint_as_float(k2);
}

// ============================ host ============================

extern "C" void kernel_launch(void* const* d_in, const int* in_sizes, int n_in,
                              void* d_out, int out_size, void* d_ws, size_t ws_size,
                              hipStream_t stream) {
  (void)in_sizes; (void)n_in; (void)out_size; (void)ws_size;
  const float* x = (const float*)d_in[0];
  char* ws = (char*)d_ws;

  u32* mm = (u32*)ws;                                   // B*6 ordered-uint min/max
  float* prm = (float*)(ws + 4096);                     // B*4: m0,m1,m2,den
  u32* counts = (u32*)(ws + 8192);                      // B*256*NBLK  (8 MB)
  u32* keysA = (u32*)(ws + 8192 + (size_t)B_ * RADIX * NBLK * 4);  // 3*B*N (100 MB)

  const size_t BN = (size_t)B_ * N_;
  u32 *A0 = keysA, *A1 = keysA + BN, *A2 = keysA + 2 * BN;
  u32* Bb = (u32*)d_out;  // d_out doubles as SoA ping-pong buffer until the last pass
  u32 *B0 = Bb, *B1 = Bb + BN, *B2 = Bb + 2 * BN;

  dim3 gridR(64, B_), gridT(NBLK, B_);

  init_mm_kernel<<<1, 256, 0, stream>>>(mm);
  reduce_mm_kernel<<<gridR, 256, 0, stream>>>(x, mm);
  params_kernel<<<1, 64, 0, stream>>>(mm, prm);

  // pass 0: x -> A, digit = byte0 of z-key
  hist_first_kernel<<<gridT, 256, 0, stream>>>(x, prm, counts);
  scan_kernel<<<B_, 256, 0, stream>>>(counts);
  scatter_first_kernel<<<gridT, 1024, 32 * RADIX * 4, stream>>>(x, prm, counts, A0, A1, A2);

  // passes 1..11 (LSD over 96-bit key: z bytes 1-3, y bytes 0-3, x bytes 0-3)
  for (int p = 1; p < 12; ++p) {
    const int w = 2 - (p >> 2);      // key word holding this pass's digit (0=x,1=y,2=z)
    const int shift = 8 * (p & 3);
    u32 *s0, *s1, *s2, *t0, *t1, *t2;
    if (p & 1) { s0 = A0; s1 = A1; s2 = A2; t0 = B0; t1 = B1; t2 = B2; }
    else       { s0 = B0; s1 = B1; s2 = B2; t0 = A0; t1 = A1; t2 = A2; }
    const u32* sw = (w == 0) ? s0 : ((w == 1) ? s1 : s2);

    hist_mid_kernel<<<gridT, 256, 0, stream>>>(sw, shift, counts);
    scan_kernel<<<B_, 256, 0, stream>>>(counts);
    const size_t shmem = (size_t)(3 * TILE + 32 * RADIX) * 4;  // 45056 B
    if (p < 11)
      scatter_mid_kernel<<<gridT, 1024, shmem, stream>>>(s0, s1, s2, t0, t1, t2, w, shift, counts);
    else
      scatter_last_kernel<<<gridT, 1024, shmem, stream>>>(s0, s1, s2, (float*)d_out, counts);
  }
}